// GeodesicAcceleration_17480516894900
// MI455X (gfx1250) — compile-verified
//
#include <hip/hip_runtime.h>
#include <hip/hip_bf16.h>

typedef __attribute__((ext_vector_type(2))) float v2f;
typedef __attribute__((ext_vector_type(4))) float v4f;
typedef __attribute__((ext_vector_type(8))) float v8f;

// out[bl, k] = - sum_{ij} Gamma[bl, k, ij] * w[ij],  w[ij] = dz[i]*dz[j]
// One (b,l) site per 64-thread block (2 wave32s). Wave w handles k in [16w, 16w+16).
// WMMA f32 16x16x4: A = Gamma tile 16 rows x 4 ij, B = w chunk broadcast over N.
__global__ __launch_bounds__(64) void GeodesicAccel_kernel(
    const float* __restrict__ dz,
    const float* __restrict__ Gamma,
    float* __restrict__ out)
{
    __shared__ float dz_s[32];
    __shared__ float w_s[1024];

    const int tid  = threadIdx.x;
    const int lane = tid & 31;
    const int bl   = blockIdx.x;

    // --- Stage dz (32 floats for this site) into LDS via CDNA5 async path ---
    if (tid < 32) {
        unsigned lds_off = (unsigned)(uintptr_t)(&dz_s[lane]);          // low 32 bits of shared ptr = LDS offset
        unsigned voff    = (unsigned)((bl * 32 + lane) * sizeof(float));
        asm volatile("global_load_async_to_lds_b32 %0, %1, %2"
                     :: "v"(lds_off), "v"(voff), "s"(dz)
                     : "memory");
        asm volatile("s_wait_asynccnt 0x0" ::: "memory");
    }
    __syncthreads();

    // --- Build the outer product w[i*32+j] = dz[i]*dz[j] in LDS ---
    for (int idx = tid; idx < 1024; idx += 64) {
        w_s[idx] = dz_s[idx >> 5] * dz_s[idx & 31];
    }
    __syncthreads();

    // --- Streaming WMMA reduction over the ij axis (K chunks of 4) ---
    const int k0   = (tid >> 5) << 4;   // wave0 -> rows 0..15, wave1 -> rows 16..31
    const int row  = lane & 15;         // M index within the 16-row tile
    const int half = lane >> 4;         // lane-half selects K={0,1} vs K={2,3}

    // A-slot (vgpr V, lane-half h) holds K = 2h+V  ->  Gamma[row, ij0 + 2h + V]
    const float* gptr = Gamma + ((size_t)bl << 15)
                              + ((size_t)(k0 + row) << 10)
                              + (size_t)(2 * half);

    v8f acc = {};     // 16x16 f32 accumulator (all N columns identical)
    int ij = 0;
    #pragma unroll 1
    for (int c = 0; c < 32; ++c) {                  // 32 chunks x 8 WMMA = 256 steps of K=4
        __builtin_prefetch(gptr + 256, 0, 0);       // 1KB ahead in this lane's row stream
        #pragma unroll
        for (int u = 0; u < 8; ++u) {
            v2f a = *(const v2f*)gptr;                       // global_load_b64 (Gamma stream)
            v2f b = *(const v2f*)(&w_s[ij + 2 * half]);      // ds_load_b64 (B broadcast over N)
            acc = __builtin_amdgcn_wmma_f32_16x16x4_f32(
                /*neg_a=*/false, a, /*neg_b=*/false, b,
                /*c_mod=*/(short)0, acc,
                /*reuse_a=*/false, /*reuse_b=*/false);
            gptr += 4;
            ij   += 4;
        }
    }

    // --- Write back column 0 of D: lane 0 holds rows k0..k0+7, lane 16 holds k0+8..k0+15 ---
    if (row == 0) {
        float* o = out + ((size_t)bl << 5) + (size_t)(k0 + (half << 3));
        v4f lo = { -acc[0], -acc[1], -acc[2], -acc[3] };
        v4f hi = { -acc[4], -acc[5], -acc[6], -acc[7] };
        *(v4f*)(o)     = lo;
        *(v4f*)(o + 4) = hi;
    }
}

extern "C" void kernel_launch(void* const* d_in, const int* in_sizes, int n_in,
                              void* d_out, int out_size, void* d_ws, size_t ws_size,
                              hipStream_t stream) {
    const float* dz    = (const float*)d_in[0];   // [B, L, 32]
    const float* Gamma = (const float*)d_in[1];   // [B, L, 32, 32, 32]
    float*       out   = (float*)d_out;           // [B, L, 32]

    const int nBL = in_sizes[0] / 32;             // B*L = 8192
    GeodesicAccel_kernel<<<dim3(nBL), dim3(64), 0, stream>>>(dz, Gamma, out);
}